// GraphWaveNet_36447092474006
// MI455X (gfx1250) — compile-verified
//
#include <hip/hip_runtime.h>
#include <math.h>

// Problem constants (from reference)
#define BB 4
#define TT 12
#define NN 10000
#define FIN 2
#define HH 64
#define EE 80000
#define M_ROWS (BB * NN)          // 40000 = 625 blocks * 4 waves * 16 rows

typedef float v2f __attribute__((ext_vector_type(2)));
typedef float v8f __attribute__((ext_vector_type(8)));

// ---------------------------------------------------------------------------
// Degree kernels: dinv[n] = rsqrt(1 + #edges with dst==n)
// ---------------------------------------------------------------------------
__global__ void k_deg_init(float* dinv) {
    int n = blockIdx.x * blockDim.x + threadIdx.x;
    if (n < NN) dinv[n] = 1.0f;                  // self loop
}

__global__ void k_deg_acc(const int* __restrict__ ei, float* dinv) {
    int e = blockIdx.x * blockDim.x + threadIdx.x;
    if (e < EE) atomicAdd(&dinv[ei[EE + e]], 1.0f);   // dst row of edge_index
}

__global__ void k_deg_fin(float* dinv) {
    int n = blockIdx.x * blockDim.x + threadIdx.x;
    if (n < NN) dinv[n] = rsqrtf(dinv[n]);
}

// ---------------------------------------------------------------------------
// Pack w2 (O=64, I=64, 1, K=3) -> P[(k*64+c)][o], k in {0,1} (tap 2 hits pad)
// ---------------------------------------------------------------------------
__global__ void k_pack_w2(const float* __restrict__ w2, float* __restrict__ P) {
    int tid = blockIdx.x * blockDim.x + threadIdx.x;   // 128*64
    if (tid >= 128 * 64) return;
    int o = tid & 63;
    int kc = tid >> 6;            // k*64 + c
    int c = kc & 63;
    int k = kc >> 6;
    P[(size_t)kc * 64 + o] = w2[(size_t)o * (64 * 3) + c * 3 + k];
}

// ---------------------------------------------------------------------------
// conv1 + ReLU, only t in {10,11}; h1[(b*N+n)*128 + k*64 + c]
// ---------------------------------------------------------------------------
__global__ void k_conv1(const float* __restrict__ x, const float* __restrict__ w1,
                        const float* __restrict__ b1, float* __restrict__ h1) {
    int tid = blockIdx.x * blockDim.x + threadIdx.x;
    if (tid >= M_ROWS * 128) return;
    int c   = tid & 63;
    int k   = (tid >> 6) & 1;     // time tap offset: t = 10 + k
    int row = tid >> 7;           // b*N + n
    int b   = row / NN;
    int n   = row - b * NN;
    float acc = b1[c];
    #pragma unroll
    for (int j = 0; j < 3; ++j) {
        int t = 9 + k + j;        // x time index (cross-correlation, pad 1)
        if (t >= TT) continue;    // zero padding at t == 12
        const float* xp = x + (((size_t)b * TT + t) * NN + n) * FIN;
        acc += xp[0] * w1[c * 6 + 0 * 3 + j];
        acc += xp[1] * w1[c * 6 + 1 * 3 + j];
    }
    h1[tid] = fmaxf(acc, 0.0f);
}

// ---------------------------------------------------------------------------
// WMMA f32 GEMM (fully specialized): C[M x 64] = act(A[M x K] @ W[K x 64] + bias)
// One wave -> 16 rows x 64 cols (4 col-tiles), V_WMMA_F32_16X16X4_F32.
// Weight panel staged into LDS pair-interleaved: sW[(k/2)*RS + col*2 + (k&1)],
// so each B fragment is one contiguous 8B -> single ds_load_b64, no shuffles.
// Pair-row stride 160 (mod 64 == 32) keeps the two half-waves on disjoint banks.
// Grid must be exactly M/64 blocks of 128 threads (no divergence; EXEC all-1s).
// ---------------------------------------------------------------------------
template <int K, int RELU_A, int RELU_OUT, int HAS_BIAS>
__global__ __launch_bounds__(128)
void k_gemm_wmma(const float* __restrict__ A, const float* __restrict__ W,
                 const float* __restrict__ bias, float* __restrict__ C) {
    constexpr int PAIRS = K / 2;
    constexpr int RS    = 160;                     // floats per pair-row (128 data + 32 pad)
    __shared__ float sW[PAIRS * RS];

    // Cooperative stage: thread handles (p, o): reads W[2p][o], W[2p+1][o],
    // writes them adjacent at sW[p*RS + o*2]. Global reads coalesced per row.
    #pragma unroll
    for (int i = 0; i < PAIRS * 64 / 128; ++i) {
        const int idx = i * 128 + threadIdx.x;     // p*64 + o
        const int p = idx >> 6;
        const int o = idx & 63;
        v2f w;
        w.x = W[(size_t)(2 * p)     * 64 + o];
        w.y = W[(size_t)(2 * p + 1) * 64 + o];
        *(v2f*)(sW + p * RS + o * 2) = w;
    }
    __syncthreads();

    const int lane = threadIdx.x & 31;
    const int wave = threadIdx.x >> 5;
    const int row0 = (blockIdx.x * 4 + wave) * 16;
    const int rm = lane & 15;                      // row within A tile / col within B tile
    const int kh = lane >> 4;                      // K half-select (0: K+0..1, 1: K+2..3)

    const float* Arow = A + (size_t)(row0 + rm) * K;
    v8f acc0 = {}, acc1 = {}, acc2 = {}, acc3 = {};

    #pragma unroll 8
    for (int kk = 0; kk < K; kk += 4) {
        const int ka = kk + 2 * kh;                // == 2*p
        v2f a = *(const v2f*)(Arow + ka);          // 8B aligned (ka even)
        if (RELU_A) { a.x = fmaxf(a.x, 0.0f); a.y = fmaxf(a.y, 0.0f); }

        const float* wp = sW + ((kk >> 1) + kh) * RS + rm * 2;
        v2f b0 = *(const v2f*)(wp +  0);           // col tile 0: cols  0..15
        v2f b1 = *(const v2f*)(wp + 32);           // col tile 1: cols 16..31
        v2f b2 = *(const v2f*)(wp + 64);           // col tile 2: cols 32..47
        v2f b3 = *(const v2f*)(wp + 96);           // col tile 3: cols 48..63

        acc0 = __builtin_amdgcn_wmma_f32_16x16x4_f32(false, a, false, b0, (short)0, acc0, false, false);
        acc1 = __builtin_amdgcn_wmma_f32_16x16x4_f32(false, a, false, b1, (short)0, acc1, false, false);
        acc2 = __builtin_amdgcn_wmma_f32_16x16x4_f32(false, a, false, b2, (short)0, acc2, false, false);
        acc3 = __builtin_amdgcn_wmma_f32_16x16x4_f32(false, a, false, b3, (short)0, acc3, false, false);
    }

    const float bc0 = HAS_BIAS ? bias[ 0 + rm] : 0.0f;
    const float bc1 = HAS_BIAS ? bias[16 + rm] : 0.0f;
    const float bc2 = HAS_BIAS ? bias[32 + rm] : 0.0f;
    const float bc3 = HAS_BIAS ? bias[48 + rm] : 0.0f;

    #pragma unroll
    for (int r = 0; r < 8; ++r) {
        const int row = row0 + r + 8 * kh;         // C/D layout: lanes>=16 hold M=r+8
        float* crow = C + (size_t)row * 64;
        float v0 = acc0[r] + bc0, v1 = acc1[r] + bc1;
        float v2 = acc2[r] + bc2, v3 = acc3[r] + bc3;
        if (RELU_OUT) {
            v0 = fmaxf(v0, 0.0f); v1 = fmaxf(v1, 0.0f);
            v2 = fmaxf(v2, 0.0f); v3 = fmaxf(v3, 0.0f);
        }
        crow[ 0 + rm] = v0;
        crow[16 + rm] = v1;
        crow[32 + rm] = v2;
        crow[48 + rm] = v3;
    }
}

// ---------------------------------------------------------------------------
// GCN aggregation: g = t * dinv[n]^2 + bias  (self loop), then edge atomics
// ---------------------------------------------------------------------------
__global__ void k_agg_init(const float* __restrict__ t, const float* __restrict__ dinv,
                           const float* __restrict__ gb, float* __restrict__ g) {
    int tid = blockIdx.x * blockDim.x + threadIdx.x;
    if (tid >= M_ROWS * HH) return;
    int c   = tid & 63;
    int row = tid >> 6;
    int n   = row % NN;
    float di = dinv[n];
    g[tid] = t[tid] * (di * di) + gb[c];
}

__global__ void k_agg_edges(const float* __restrict__ t, const int* __restrict__ ei,
                            const float* __restrict__ dinv, float* __restrict__ g) {
    long long tid = (long long)blockIdx.x * blockDim.x + threadIdx.x;
    if (tid >= (long long)BB * EE * HH) return;
    int c    = (int)(tid & 63);
    long long r = tid >> 6;
    int e    = (int)(r % EE);
    int b    = (int)(r / EE);
    int src  = ei[e];
    int dst  = ei[EE + e];
    float norm = dinv[src] * dinv[dst];
    float val  = t[((size_t)b * NN + src) * 64 + c] * norm;
    atomicAdd(&g[((size_t)b * NN + dst) * 64 + c], val);
}

// ---------------------------------------------------------------------------
// Output: out[b,n] = bo + sum_c relu(a[row,c]) * wo[c]
// ---------------------------------------------------------------------------
__global__ void k_out(const float* __restrict__ a, const float* __restrict__ wo,
                      const float* __restrict__ bo, float* __restrict__ out) {
    int row = blockIdx.x * blockDim.x + threadIdx.x;   // b*N + n
    if (row >= M_ROWS) return;
    const float* ar = a + (size_t)row * 64;
    float acc = bo[0];
    #pragma unroll
    for (int c = 0; c < 64; ++c)
        acc += fmaxf(ar[c], 0.0f) * wo[c];
    out[row] = acc;
}

// ---------------------------------------------------------------------------
// Launch
// ---------------------------------------------------------------------------
extern "C" void kernel_launch(void* const* d_in, const int* in_sizes, int n_in,
                              void* d_out, int out_size, void* d_ws, size_t ws_size,
                              hipStream_t stream) {
    const float* x   = (const float*)d_in[0];
    const int*   ei  = (const int*)  d_in[1];
    // d_in[2] = edge_attr (unused by reference)
    const float* w1  = (const float*)d_in[3];
    const float* b1  = (const float*)d_in[4];
    const float* w2  = (const float*)d_in[5];
    const float* b2  = (const float*)d_in[6];
    const float* gw1 = (const float*)d_in[7];
    const float* gb1 = (const float*)d_in[8];
    const float* gw2 = (const float*)d_in[9];
    const float* gb2 = (const float*)d_in[10];
    const float* wo  = (const float*)d_in[11];
    const float* bo  = (const float*)d_in[12];
    float* out = (float*)d_out;

    // Workspace layout (floats), total ~51.3 MB
    float* ws = (float*)d_ws;
    float* dinv = ws;                               // 10240
    float* P    = ws + 10240;                       // 8192  (packed w2, [128][64])
    float* h1   = ws + 18432;                       // 5,120,000  [M][128]
    float* tb   = h1 + (size_t)M_ROWS * 128;        // 2,560,000  matmul results
    float* gb_  = tb + (size_t)M_ROWS * 64;         // 2,560,000  aggregation results
    float* ab   = gb_ + (size_t)M_ROWS * 64;        // 2,560,000  gcn input / output ping
    (void)ws_size; (void)in_sizes; (void)n_in; (void)out_size;

    const int TPB = 256;

    // degrees
    k_deg_init<<<(NN + TPB - 1) / TPB, TPB, 0, stream>>>(dinv);
    k_deg_acc <<<(EE + TPB - 1) / TPB, TPB, 0, stream>>>(ei, dinv);
    k_deg_fin <<<(NN + TPB - 1) / TPB, TPB, 0, stream>>>(dinv);

    // pack conv2 weights
    k_pack_w2<<<(128 * 64 + TPB - 1) / TPB, TPB, 0, stream>>>(w2, P);

    // conv1 + relu at t in {10,11}
    k_conv1<<<(M_ROWS * 128 + TPB - 1) / TPB, TPB, 0, stream>>>(x, w1, b1, h1);

    const int gemmBlocks = M_ROWS / 64;   // 625, exact

    // conv2 at t=11: relu(h1 @ P + b2) -> ab
    k_gemm_wmma<128, 0, 1, 1><<<gemmBlocks, 128, 0, stream>>>(h1, P, b2, ab);

    // GCN layer 1: tb = ab @ gw1 ; gb_ = scatter(tb) + self + gb1
    k_gemm_wmma<64, 0, 0, 0><<<gemmBlocks, 128, 0, stream>>>(ab, gw1, nullptr, tb);
    k_agg_init<<<(M_ROWS * HH + TPB - 1) / TPB, TPB, 0, stream>>>(tb, dinv, gb1, gb_);
    {
        long long tot = (long long)BB * EE * HH;
        k_agg_edges<<<(unsigned)((tot + TPB - 1) / TPB), TPB, 0, stream>>>(tb, ei, dinv, gb_);
    }

    // GCN layer 2: tb = relu(gb_) @ gw2 ; ab = scatter(tb) + self + gb2
    k_gemm_wmma<64, 1, 0, 0><<<gemmBlocks, 128, 0, stream>>>(gb_, gw2, nullptr, tb);
    k_agg_init<<<(M_ROWS * HH + TPB - 1) / TPB, TPB, 0, stream>>>(tb, dinv, gb2, ab);
    {
        long long tot = (long long)BB * EE * HH;
        k_agg_edges<<<(unsigned)((tot + TPB - 1) / TPB), TPB, 0, stream>>>(tb, ei, dinv, ab);
    }

    // output conv (1x1) + pick t=T-1 (already the only t we computed)
    k_out<<<(M_ROWS + TPB - 1) / TPB, TPB, 0, stream>>>(ab, wo, bo, out);
}